// StationGNN_35459249996283
// MI455X (gfx1250) — compile-verified
//
#include <hip/hip_runtime.h>

// ---------------------------------------------------------------------------
// GraphSAGE (3 layers, HIDDEN=256) + MLP head for MI455X (gfx1250, wave32).
// fp32 WMMA (v_wmma_f32_16x16x4_f32) keeps full precision; scatter-mean is the
// real bottleneck and is L2-resident (102 MB feature matrix < 192 MB L2).
// Weights are pre-packed into WMMA B-fragment order so every B fragment is one
// coalesced b64 load; accumulators are register-blocked 8 tiles at a time to
// stay under 256 VGPRs (no s_set_vgpr_msb in the hot loop).
// ---------------------------------------------------------------------------

typedef __attribute__((ext_vector_type(2))) float v2f;
typedef __attribute__((ext_vector_type(8))) float v8f;

#define N_NODES 100000
#define N_EDGES 1600000
#define HID     256
#define KC      64      // K-chunk staged in LDS per wave (16 x 64 f32 = 4 KB)

// ---------------------------------------------------------------- utilities
__global__ __launch_bounds__(256) void fill0_kernel(float4* p, int n4) {
    int i = blockIdx.x * blockDim.x + threadIdx.x;
    if (i < n4) p[i] = make_float4(0.f, 0.f, 0.f, 0.f);
}

__global__ __launch_bounds__(256) void count_edges_kernel(const int* __restrict__ dst,
                                                          float* __restrict__ cnt, int ne) {
    int e = blockIdx.x * blockDim.x + threadIdx.x;
    if (e < ne) atomicAdd(&cnt[dst[e]], 1.0f);
}

__global__ __launch_bounds__(256) void recip_kernel(float* __restrict__ inv,
                                                    const float* __restrict__ cnt, int n) {
    int i = blockIdx.x * blockDim.x + threadIdx.x;
    if (i < n) inv[i] = 1.0f / fmaxf(cnt[i], 1.0f);
}

// scatter-add: 64 lanes per edge, 4 floats per lane (float4 gather, f32 atomics)
__global__ __launch_bounds__(256) void scatter_add_kernel(const float* __restrict__ x,
                                                          const int* __restrict__ src,
                                                          const int* __restrict__ dst,
                                                          float* __restrict__ agg, int ne) {
    long long t = (long long)blockIdx.x * blockDim.x + threadIdx.x;
    int e  = (int)(t >> 6);
    int c4 = (int)(t & 63) << 2;
    if (e >= ne) return;
    int s = src[e], d = dst[e];
    float4 v = *(const float4*)(x + (size_t)s * HID + c4);
    float* o = agg + (size_t)d * HID + c4;
    atomicAdd(o + 0, v.x);
    atomicAdd(o + 1, v.y);
    atomicAdd(o + 2, v.z);
    atomicAdd(o + 3, v.w);
}

// ---------------------------------------------------------------------------
// Repack row-major W [256 x N] into WMMA B-fragment order (K fixed at 256):
//   float index t = ((nt*64 + ks)*32 + lane)*2 + j
//   maps to      W[(4*ks + 2*(lane>>4) + j) * N + nt*16 + (lane&15)]
// so the GEMM reads one contiguous v2f per lane per fragment.
// ---------------------------------------------------------------------------
__global__ __launch_bounds__(256) void pack_b_kernel(const float* __restrict__ W,
                                                     float* __restrict__ P, int N) {
    int t = blockIdx.x * blockDim.x + threadIdx.x;
    if (t >= 256 * N) return;
    int j    = t & 1;
    int lane = (t >> 1) & 31;
    int ks   = (t >> 6) & 63;
    int nt   = t >> 12;
    int k = (ks << 2) + ((lane >> 4) << 1) + j;
    int n = (nt << 4) + (lane & 15);
    P[t] = W[k * N + n];
}

// ---------------------------------------------------------------------------
// Fused SAGE layer: out = relu( (agg*inv) @ Wl + bl + x @ Wr )
// 8 waves/block, each wave owns 16 rows. N-tiles processed in 2 groups of 8
// (64 accumulator VGPRs live). A staged in per-wave LDS; B from packed
// fragment stream (coalesced b64 per lane).
// WMMA layouts per CDNA5 ISA 7.12.2:
//   A 16x4 : lanes 0-15 -> (K, K+1), lanes 16-31 -> (K+2, K+3), row = lane&15
//   B 4x16 : VGPR0/1 rows K+koff / K+koff+1, col = n0 + (lane&15)
//   C 16x16: VGPR i -> row i (lanes 0-15) / row i+8 (lanes 16-31)
// ---------------------------------------------------------------------------
__global__ __launch_bounds__(256) void sage_wmma_kernel(
    const float* __restrict__ agg, const float* __restrict__ invc,
    const float* __restrict__ x,
    const v2f* __restrict__ PWl, const float* __restrict__ bl,
    const v2f* __restrict__ PWr,
    float* __restrict__ out, int n_nodes)
{
    __shared__ float lds[8 * 16 * KC];              // 32 KB total
    const int wave = threadIdx.x >> 5;
    const int lane = threadIdx.x & 31;
    const int row0 = (blockIdx.x * 8 + wave) * 16;
    if (row0 >= n_nodes) return;                    // wave-uniform (no block syncs used)

    float* As = lds + wave * 16 * KC;
    const int m    = lane & 15;
    const int koff = (lane >> 4) << 1;              // 0 for lanes 0-15, 2 for 16-31

    for (int g = 0; g < 2; ++g) {                   // N-tile group: tiles g*8 .. g*8+7
        v8f c[8];
        #pragma unroll
        for (int nt = 0; nt < 8; ++nt) {
            const float bv = bl[((g * 8 + nt) << 4) + m];
            #pragma unroll
            for (int i = 0; i < 8; ++i) c[nt][i] = bv;
        }

        #pragma unroll
        for (int phase = 0; phase < 2; ++phase) {
            const float* __restrict__ A = phase ? x   : agg;
            const v2f*   __restrict__ P = phase ? PWr : PWl;
            for (int kc = 0; kc < HID; kc += KC) {
                // stage 16 x KC chunk of A (scaled by 1/deg in phase 0), coalesced
                #pragma unroll
                for (int t = lane; t < 16 * (KC / 4); t += 32) {
                    const int r  = t >> 4;          // KC/4 == 16 float4 per row
                    const int c4 = (t & 15) << 2;
                    float4 v = *(const float4*)(A + (size_t)(row0 + r) * HID + kc + c4);
                    float s = (phase == 0) ? invc[row0 + r] : 1.0f;
                    *(float4*)(As + r * KC + c4) = make_float4(v.x * s, v.y * s, v.z * s, v.w * s);
                }
                // consume: 8 N-tiles x (KC/4) wmma steps, B streamed from packed layout
                #pragma unroll
                for (int nt = 0; nt < 8; ++nt) {
                    const v2f* __restrict__ Bn =
                        P + ((size_t)(g * 8 + nt) * 64 + (kc >> 2)) * 32 + lane;
                    const float* __restrict__ Arow = As + m * KC;
                    #pragma unroll 4
                    for (int ks = 0; ks < KC / 4; ++ks) {
                        const int ka = (ks << 2) + koff;
                        v2f a; a.x = Arow[ka]; a.y = Arow[ka + 1];
                        v2f b = Bn[ks * 32];
                        c[nt] = __builtin_amdgcn_wmma_f32_16x16x4_f32(
                            false, a, false, b, (short)0, c[nt], false, false);
                    }
                }
            }
        }

        const int mbase = row0 + ((lane >> 4) << 3);
        #pragma unroll
        for (int nt = 0; nt < 8; ++nt) {
            const int col = ((g * 8 + nt) << 4) + m;
            #pragma unroll
            for (int i = 0; i < 8; ++i) {
                float v = c[nt][i];
                out[(size_t)(mbase + i) * HID + col] = v > 0.0f ? v : 0.0f;
            }
        }
    }
}

// ---------------------------------------------------------------------------
// MLP layer 1: h = relu( x @ W1 + b1 ),  W1 packed, [256,128]
// ---------------------------------------------------------------------------
__global__ __launch_bounds__(256) void mlp1_wmma_kernel(
    const float* __restrict__ x,
    const v2f* __restrict__ PW1, const float* __restrict__ b1,
    float* __restrict__ h, int n_nodes)
{
    __shared__ float lds[8 * 16 * KC];
    const int wave = threadIdx.x >> 5;
    const int lane = threadIdx.x & 31;
    const int row0 = (blockIdx.x * 8 + wave) * 16;
    if (row0 >= n_nodes) return;

    float* As = lds + wave * 16 * KC;
    const int m    = lane & 15;
    const int koff = (lane >> 4) << 1;
    const int LDB  = 128;

    v8f c[8];
    #pragma unroll
    for (int nt = 0; nt < 8; ++nt) {
        const float bv = b1[(nt << 4) + m];
        #pragma unroll
        for (int i = 0; i < 8; ++i) c[nt][i] = bv;
    }

    for (int kc = 0; kc < HID; kc += KC) {
        #pragma unroll
        for (int t = lane; t < 16 * (KC / 4); t += 32) {
            const int r  = t >> 4;
            const int c4 = (t & 15) << 2;
            *(float4*)(As + r * KC + c4) =
                *(const float4*)(x + (size_t)(row0 + r) * HID + kc + c4);
        }
        #pragma unroll
        for (int nt = 0; nt < 8; ++nt) {
            const v2f* __restrict__ Bn = PW1 + ((size_t)nt * 64 + (kc >> 2)) * 32 + lane;
            const float* __restrict__ Arow = As + m * KC;
            #pragma unroll 4
            for (int ks = 0; ks < KC / 4; ++ks) {
                const int ka = (ks << 2) + koff;
                v2f a; a.x = Arow[ka]; a.y = Arow[ka + 1];
                v2f b = Bn[ks * 32];
                c[nt] = __builtin_amdgcn_wmma_f32_16x16x4_f32(
                    false, a, false, b, (short)0, c[nt], false, false);
            }
        }
    }

    const int mbase = row0 + ((lane >> 4) << 3);
    #pragma unroll
    for (int nt = 0; nt < 8; ++nt) {
        const int col = (nt << 4) + m;
        #pragma unroll
        for (int i = 0; i < 8; ++i) {
            float v = c[nt][i];
            h[(size_t)(mbase + i) * LDB + col] = v > 0.0f ? v : 0.0f;
        }
    }
}

// ---------------------------------------------------------------------------
// MLP layer 2 (head): out = h @ W2 + b2,  W2: [128,4]; one thread per node.
// ---------------------------------------------------------------------------
__global__ __launch_bounds__(256) void mlp2_kernel(
    const float* __restrict__ h, const float* __restrict__ W2,
    const float* __restrict__ b2, float* __restrict__ out, int n_nodes)
{
    __shared__ float w[128 * 4];
    __shared__ float bb[4];
    for (int t = threadIdx.x; t < 512; t += 256) w[t] = W2[t];
    if (threadIdx.x < 4) bb[threadIdx.x] = b2[threadIdx.x];
    __syncthreads();

    int i = blockIdx.x * blockDim.x + threadIdx.x;
    if (i >= n_nodes) return;
    float a0 = bb[0], a1 = bb[1], a2 = bb[2], a3 = bb[3];
    const float* hr = h + (size_t)i * 128;
    #pragma unroll 8
    for (int k = 0; k < 128; ++k) {
        float hv = hr[k];
        a0 += hv * w[k * 4 + 0];
        a1 += hv * w[k * 4 + 1];
        a2 += hv * w[k * 4 + 2];
        a3 += hv * w[k * 4 + 3];
    }
    float* o = out + (size_t)i * 4;
    o[0] = a0; o[1] = a1; o[2] = a2; o[3] = a3;
}

// ---------------------------------------------------------------------------
extern "C" void kernel_launch(void* const* d_in, const int* in_sizes, int n_in,
                              void* d_out, int out_size, void* d_ws, size_t ws_size,
                              hipStream_t stream) {
    const float* x0 = (const float*)d_in[0];
    const int*   ei = (const int*)d_in[1];
    const float* Wl = (const float*)d_in[2];   // [3,256,256]
    const float* bl = (const float*)d_in[3];   // [3,256]
    const float* Wr = (const float*)d_in[4];   // [3,256,256]
    const float* W1 = (const float*)d_in[5];   // [256,128]
    const float* b1 = (const float*)d_in[6];   // [128]
    const float* W2 = (const float*)d_in[7];   // [128,4]
    const float* b2 = (const float*)d_in[8];   // [4]
    float* out = (float*)d_out;

    float* ws  = (float*)d_ws;
    float* AGG = ws;                            // 25.6M f32
    float* XA  = AGG + (size_t)N_NODES * HID;   // 25.6M f32
    float* XB  = XA  + (size_t)N_NODES * HID;   // 25.6M f32 (also reused as H)
    float* CNT = XB  + (size_t)N_NODES * HID;   // 100K f32
    float* INV = CNT + N_NODES;                 // 100K f32
    float* PWl = INV + N_NODES;                 // 3 x 64K f32 packed Wl
    float* PWr = PWl + 3 * HID * HID;           // 3 x 64K f32 packed Wr
    float* PW1 = PWr + 3 * HID * HID;           // 32K f32 packed W1
    float* H   = XB;                            // reuse XB after layer 2

    const int* src = ei;
    const int* dst = ei + N_EDGES;

    // pack weights into WMMA B-fragment order (cheap, once per launch)
    for (int l = 0; l < 3; ++l) {
        pack_b_kernel<<<256, 256, 0, stream>>>(Wl + (size_t)l * HID * HID,
                                               PWl + (size_t)l * HID * HID, HID);
        pack_b_kernel<<<256, 256, 0, stream>>>(Wr + (size_t)l * HID * HID,
                                               PWr + (size_t)l * HID * HID, HID);
    }
    pack_b_kernel<<<128, 256, 0, stream>>>(W1, PW1, 128);

    // in-degree counts -> 1/max(cnt,1)
    fill0_kernel<<<(N_NODES / 4 + 255) / 256, 256, 0, stream>>>((float4*)CNT, N_NODES / 4);
    count_edges_kernel<<<(N_EDGES + 255) / 256, 256, 0, stream>>>(dst, CNT, N_EDGES);
    recip_kernel<<<(N_NODES + 255) / 256, 256, 0, stream>>>(INV, CNT, N_NODES);

    const int agg4 = N_NODES * HID / 4;                         // 6.4M float4
    const int scatter_blocks = (int)(((long long)N_EDGES * 64 + 255) / 256);
    const int gemm_blocks = (N_NODES + 127) / 128;

    const float* xin = x0;
    float* louts[3] = {XA, XB, XA};
    for (int l = 0; l < 3; ++l) {
        fill0_kernel<<<(agg4 + 255) / 256, 256, 0, stream>>>((float4*)AGG, agg4);
        scatter_add_kernel<<<scatter_blocks, 256, 0, stream>>>(xin, src, dst, AGG, N_EDGES);
        sage_wmma_kernel<<<gemm_blocks, 256, 0, stream>>>(
            AGG, INV, xin,
            (const v2f*)(PWl + (size_t)l * HID * HID), bl + l * HID,
            (const v2f*)(PWr + (size_t)l * HID * HID), louts[l], N_NODES);
        xin = louts[l];
    }

    mlp1_wmma_kernel<<<gemm_blocks, 256, 0, stream>>>(xin, (const v2f*)PW1, b1, H, N_NODES);
    mlp2_kernel<<<(N_NODES + 255) / 256, 256, 0, stream>>>(H, W2, b2, out, N_NODES);
}